// EquivariantLocalScoreMachine_49460843381565
// MI455X (gfx1250) — compile-verified
//
#include <hip/hip_runtime.h>
#include <hip/hip_bf16.h>

typedef __attribute__((ext_vector_type(2))) float v2f;
typedef __attribute__((ext_vector_type(8))) float v8f;

#define Bq    2
#define Cc    3
#define Hh    32
#define Ww    32
#define HW    1024
#define NIMG  16
#define PTOT  32768          // total patches over both scan chunks
#define K27   27
#define KP    28             // padded K
#define NTOT  2048           // B * HW columns
#define BM    128            // M rows per block iteration
#define MSPLIT 64
#define MITER (PTOT / (BM * MSPLIT))   // 4

__device__ __forceinline__ unsigned fkey(float f) {
    unsigned u = __float_as_uint(f);
    return (u & 0x80000000u) ? ~u : (u | 0x80000000u);
}
__device__ __forceinline__ float fdec(unsigned k) {
    return __uint_as_float((k & 0x80000000u) ? (k & 0x7fffffffu) : ~k);
}

// Build A[PTOT][KP] (zero-padded 3x3 patches), per-patch norms, centers.
__global__ void k_prep_patches(const float* __restrict__ images,
                               float* __restrict__ A, float* __restrict__ PN,
                               float* __restrict__ PC) {
    int p = blockIdx.x * blockDim.x + threadIdx.x;
    if (p >= PTOT) return;
    int s = p >> 14, rem = p & 16383;
    int n = rem >> 10, hw = rem & 1023;
    int h = hw >> 5, w = hw & 31;
    const float* img = images + (size_t)((s * NIMG + n) * Cc) * HW;
    float pn = 0.f;
#pragma unroll
    for (int c = 0; c < Cc; ++c) {
#pragma unroll
        for (int kh = 0; kh < 3; ++kh) {
#pragma unroll
            for (int kw = 0; kw < 3; ++kw) {
                int hh = h + kh - 1, ww = w + kw - 1;
                float v = 0.f;
                if (hh >= 0 && hh < Hh && ww >= 0 && ww < Ww)
                    v = img[c * HW + hh * Ww + ww];
                A[(size_t)p * KP + (c * 9 + kh * 3 + kw)] = v;
                pn += v * v;
            }
        }
        PC[p * 3 + c] = img[c * HW + h * Ww + w];   // center (kh=kw=1)
    }
    A[(size_t)p * KP + K27] = 0.f;
    PN[p] = pn;
}

// Build im2col of wrap-padded x, x-window-sum-squared, and init accumulators.
__global__ void k_prep_x(const float* __restrict__ x, float* __restrict__ XC,
                         float* __restrict__ XN, unsigned* __restrict__ MK,
                         float* __restrict__ SW, float* __restrict__ SD) {
    int t = blockIdx.x * blockDim.x + threadIdx.x;
    if (t >= NTOT) return;
    int b = t >> 10, hw = t & 1023;
    int h = hw >> 5, w = hw & 31;
    const float* xb = x + b * Cc * HW;
    float ssum = 0.f;
#pragma unroll
    for (int c = 0; c < Cc; ++c) {
#pragma unroll
        for (int kh = 0; kh < 3; ++kh) {
#pragma unroll
            for (int kw = 0; kw < 3; ++kw) {
                int hh = (h + kh - 1) & 31;           // wrap
                int ww = (w + kw - 1) & 31;
                float v = xb[c * HW + hh * Ww + ww];
                XC[(b * KP + (c * 9 + kh * 3 + kw)) * HW + hw] = v;
                ssum += v;
            }
        }
    }
    XC[(b * KP + K27) * HW + hw] = 0.f;
    XN[t] = ssum * ssum;
    SW[t] = 0.f;
    if (t < HW) MK[t] = 0u;
    SD[t] = 0.f; SD[t + NTOT] = 0.f; SD[t + 2 * NTOT] = 0.f;
}

// PASS 1: global per-(h,w) max of w.  PASS 2: softmax sums.
// conv = A(PTOT x KP) * Xcol(KP x NTOT) via V_WMMA_F32_16X16X4_F32.
template <int PASS>
__global__ __launch_bounds__(256)
void k_gemm(const float* __restrict__ A, const float* __restrict__ PN,
            const float* __restrict__ PC, const float* __restrict__ XC,
            const float* __restrict__ XN, unsigned* __restrict__ MK,
            float* __restrict__ SW, float* __restrict__ SD,
            const float* __restrict__ mu, const float* __restrict__ sigma,
            const int* __restrict__ tptr) {
    __shared__ float Bs[KP][16];
    __shared__ float As[BM][KP];
    __shared__ float red[8][16][4];

    const int tid  = threadIdx.x;
    const int wave = tid >> 5;
    const int lane = tid & 31;
    const int nloc = lane & 15;
    const int hi   = lane >> 4;      // 0: K pair {0,1}; 1: K pair {2,3}
    const int koff = hi << 1;

    const int t = *tptr;
    const float mu_t = mu[t];
    const float sg = sigma[t];
    const float inv2s2 = 1.0f / (2.0f * sg * sg);
    const float mu2 = mu_t * mu_t;

    const int nTile = blockIdx.x;            // 128 column tiles
    const int nglob = nTile * 16 + nloc;
    const int hw = nglob & 1023;

    // B tile (KP x 16), fixed for whole block
    for (int i = tid; i < KP * 16; i += 256) {
        int k = i >> 4, n = i & 15;
        int ng = nTile * 16 + n;
        Bs[k][n] = XC[((ng >> 10) * KP + k) * HW + (ng & 1023)];
    }

    const float xn = XN[nglob];
    float sub = 0.f;
    if (PASS == 2) sub = fdec(MK[hw]);

    float accMax = -3.4e38f;
    float accSW = 0.f, accSD0 = 0.f, accSD1 = 0.f, accSD2 = 0.f;

    for (int it = 0; it < MITER; ++it) {
        const int mBase = (blockIdx.y * MITER + it) * BM;
        __syncthreads();                       // As reuse hazard
        const float* Ag = A + (size_t)mBase * KP;
        for (int i = tid; i < BM * KP; i += 256) As[i / KP][i % KP] = Ag[i];
        if (it + 1 < MITER) __builtin_prefetch(Ag + BM * KP + tid, 0, 1);
        __syncthreads();

        const int arow = wave * 16 + nloc;     // A-frag: lane holds row M=lane&15
        v8f c = {0.f, 0.f, 0.f, 0.f, 0.f, 0.f, 0.f, 0.f};
#pragma unroll
        for (int kk = 0; kk < KP / 4; ++kk) {
            const int kb = kk * 4 + koff;
            v2f av, bv;
            av.x = As[arow][kb];  av.y = As[arow][kb + 1];
            bv.x = Bs[kb][nloc];  bv.y = Bs[kb + 1][nloc];
            c = __builtin_amdgcn_wmma_f32_16x16x4_f32(
                    false, av, false, bv, (short)0, c, false, false);
        }

#pragma unroll
        for (int j = 0; j < 8; ++j) {          // C: VGPR j -> M=j (lo half) / j+8
            const int mg = mBase + wave * 16 + (hi << 3) + j;
            const float wv = (2.0f * mu_t * c[j] - xn - mu2 * PN[mg]) * inv2s2;
            if (PASS == 1) {
                accMax = fmaxf(accMax, wv);
            } else {
                const float e = __expf(wv - sub);
                accSW  += e;
                accSD0 += e * PC[mg * 3 + 0];
                accSD1 += e * PC[mg * 3 + 1];
                accSD2 += e * PC[mg * 3 + 2];
            }
        }
    }

    // lane <-> lane^16 share the same column
    if (PASS == 1) {
        accMax = fmaxf(accMax, __shfl_xor(accMax, 16, 32));
        if (hi == 0) red[wave][nloc][0] = accMax;
    } else {
        accSW  += __shfl_xor(accSW, 16, 32);
        accSD0 += __shfl_xor(accSD0, 16, 32);
        accSD1 += __shfl_xor(accSD1, 16, 32);
        accSD2 += __shfl_xor(accSD2, 16, 32);
        if (hi == 0) {
            red[wave][nloc][0] = accSW;  red[wave][nloc][1] = accSD0;
            red[wave][nloc][2] = accSD1; red[wave][nloc][3] = accSD2;
        }
    }
    __syncthreads();
    if (PASS == 1) {
        if (tid < 16) {
            float m = red[0][tid][0];
#pragma unroll
            for (int w2 = 1; w2 < 8; ++w2) m = fmaxf(m, red[w2][tid][0]);
            atomicMax(&MK[(nTile * 16 + tid) & 1023], fkey(m));
        }
    } else {
        if (tid < 64) {
            const int n = tid & 15, slot = tid >> 4;
            float s = red[0][n][slot];
#pragma unroll
            for (int w2 = 1; w2 < 8; ++w2) s += red[w2][n][slot];
            const int ng = nTile * 16 + n;
            const int bb = ng >> 10, hh = ng & 1023;
            if (slot == 0) atomicAdd(&SW[ng], s);
            else           atomicAdd(&SD[(bb * 3 + (slot - 1)) * HW + hh], s);
        }
    }
}

__global__ void k_final(const float* __restrict__ x, const float* __restrict__ SW,
                        const float* __restrict__ SD, const float* __restrict__ mu,
                        const float* __restrict__ sigma, const int* __restrict__ tptr,
                        float* __restrict__ out) {
    int i = blockIdx.x * blockDim.x + threadIdx.x;   // over B*C*HW
    if (i >= Bq * Cc * HW) return;
    int b = i / (Cc * HW);
    int hw = i & 1023;
    const int t = *tptr;
    const float mu_t = mu[t], sg = sigma[t];
    const float sw = SW[b * HW + hw];
    out[i] = (mu_t * SD[i] - x[i] * sw) / (sg * sg * sw);
}

extern "C" void kernel_launch(void* const* d_in, const int* in_sizes, int n_in,
                              void* d_out, int out_size, void* d_ws, size_t ws_size,
                              hipStream_t stream) {
    const float* x      = (const float*)d_in[0];
    const float* images = (const float*)d_in[1];
    const float* mu     = (const float*)d_in[2];
    const float* sigma  = (const float*)d_in[3];
    const int*   t      = (const int*)d_in[4];

    float* ws = (float*)d_ws;
    float* A  = ws;                        // PTOT*KP
    float* PN = A  + (size_t)PTOT * KP;    // PTOT
    float* PC = PN + PTOT;                 // PTOT*3
    float* XC = PC + (size_t)PTOT * 3;     // Bq*KP*HW
    float* XN = XC + (size_t)Bq * KP * HW; // NTOT
    unsigned* MK = (unsigned*)(XN + NTOT); // HW
    float* SW = (float*)(MK + HW);         // NTOT
    float* SD = SW + NTOT;                 // 3*NTOT

    k_prep_patches<<<PTOT / 256, 256, 0, stream>>>(images, A, PN, PC);
    k_prep_x<<<NTOT / 256, 256, 0, stream>>>(x, XC, XN, MK, SW, SD);
    dim3 grid(NTOT / 16, MSPLIT);
    k_gemm<1><<<grid, 256, 0, stream>>>(A, PN, PC, XC, XN, MK, SW, SD, mu, sigma, t);
    k_gemm<2><<<grid, 256, 0, stream>>>(A, PN, PC, XC, XN, MK, SW, SD, mu, sigma, t);
    k_final<<<(Bq * Cc * HW + 255) / 256, 256, 0, stream>>>(x, SW, SD, mu, sigma, t,
                                                            (float*)d_out);
}